// DGCN_87926570483749
// MI455X (gfx1250) — compile-verified
//
#include <hip/hip_runtime.h>
#include <hip/hip_bf16.h>
#include <math.h>

typedef float v2f __attribute__((ext_vector_type(2)));
typedef float v8f __attribute__((ext_vector_type(8)));

#define BATCH 16
#define NN    2048
#define CC    64
#define EE    16
#define OO    64

// D = A(16x4 f32) * B(4x16 f32) + C(16x16 f32), exact f32 matrix op on CDNA5.
__device__ __forceinline__ v8f wmma4(v2f a, v2f b, v8f c) {
  return __builtin_amdgcn_wmma_f32_16x16x4_f32(false, a, false, b, (short)0, c,
                                               false, false);
}

// ---------------------------------------------------------------------------
// Phase 1: hyper-MLP -> nodevec V[b,n,e] = tanh(emb0 * (sig(sig(x@w1+b1)@w2+b2)@w3+b3))
// One thread per (b,n). Weight reads are lane-uniform -> scalar loads.
// ---------------------------------------------------------------------------
__global__ void k_hyper(const float* __restrict__ x, const float* __restrict__ emb0,
                        const float* __restrict__ w1, const float* __restrict__ b1,
                        const float* __restrict__ w2, const float* __restrict__ b2,
                        const float* __restrict__ w3, const float* __restrict__ b3,
                        float* __restrict__ V) {
  int idx = blockIdx.x * blockDim.x + threadIdx.x;      // b*N + n   (0..32767)
  const float* xr = x + (size_t)idx * CC;

  float a1[16];
#pragma unroll
  for (int h = 0; h < 16; ++h) a1[h] = b1[h];
  for (int i = 0; i < CC; ++i) {
    float xi = xr[i];
#pragma unroll
    for (int h = 0; h < 16; ++h) a1[h] = fmaf(xi, w1[i * 16 + h], a1[h]);
  }
#pragma unroll
  for (int h = 0; h < 16; ++h) a1[h] = 1.f / (1.f + __expf(-a1[h]));

  float h2[2];
#pragma unroll
  for (int m = 0; m < 2; ++m) {
    float a = b2[m];
#pragma unroll
    for (int h = 0; h < 16; ++h) a = fmaf(a1[h], w2[h * 2 + m], a);
    h2[m] = 1.f / (1.f + __expf(-a));
  }

  const float* e0 = emb0 + (size_t)idx * EE;
  float* vo = V + (size_t)idx * EE;
#pragma unroll
  for (int d = 0; d < EE; ++d) {
    float f = b3[d] + h2[0] * w3[d] + h2[1] * w3[EE + d];
    vo[d] = tanhf(e0[d] * f);
  }
}

// ---------------------------------------------------------------------------
// Phase 2: d[b,n] = (sum_m relu(v_n . v_m))^{-1/2}
// One wave per 16-row block; S-tiles via chained wmma_f32_16x16x4 (K=16).
// ---------------------------------------------------------------------------
__global__ void k_rowsum(const float* __restrict__ V, float* __restrict__ dv) {
  int wv   = threadIdx.x >> 5;
  int lane = threadIdx.x & 31;
  int wave = blockIdx.x * 4 + wv;
  int b  = wave >> 7;                 // NN/16 = 128 row-blocks per batch
  int n0 = (wave & 127) << 4;
  int hl  = lane >> 4;                // lane half selects K pair
  int l16 = lane & 15;

  const float* Vb = V + (size_t)b * NN * EE;

  v2f a[4];                           // A = V rows n0..n0+15, K = E = 16
  const float* arow = Vb + (size_t)(n0 + l16) * EE;
#pragma unroll
  for (int kk = 0; kk < 4; ++kk) {
    int k0 = kk * 4 + hl * 2;
    a[kk].x = arow[k0]; a[kk].y = arow[k0 + 1];
  }

  float rs[8];
#pragma unroll
  for (int i = 0; i < 8; ++i) rs[i] = 0.f;

  for (int m0 = 0; m0 < NN; m0 += 16) {
    const float* brow = Vb + (size_t)(m0 + l16) * EE;   // B = V^T tile
    v8f c = {};
#pragma unroll
    for (int kk = 0; kk < 4; ++kk) {
      int k0 = kk * 4 + hl * 2;
      v2f bb; bb.x = brow[k0]; bb.y = brow[k0 + 1];
      c = wmma4(a[kk], bb, c);
    }
#pragma unroll
    for (int i = 0; i < 8; ++i) rs[i] += fmaxf(c[i], 0.f);
  }

  // reduce the 16 columns held across each lane-half
#pragma unroll
  for (int off = 1; off < 16; off <<= 1) {
#pragma unroll
    for (int i = 0; i < 8; ++i) rs[i] += __shfl_xor(rs[i], off, 32);
  }
  if (l16 == 0) {
#pragma unroll
    for (int i = 0; i < 8; ++i)
      dv[b * NN + n0 + hl * 8 + i] = 1.f / sqrtf(rs[i]);
  }
}

// ---------------------------------------------------------------------------
// Phase 3: xg2 = (d * relu(V V^T) * d) @ x, fused (never materialize adj).
// Per m-tile: S via 4 wmma, relu+scale, LDS transpose C-layout -> A-layout,
// then 16 wmma into the 16x64 accumulator.
// ---------------------------------------------------------------------------
__global__ void k_lx(const float* __restrict__ V, const float* __restrict__ dv,
                     const float* __restrict__ x, float* __restrict__ xg2) {
  __shared__ float sbuf[4][16][16];   // per-wave S tile for layout conversion

  int wv   = threadIdx.x >> 5;
  int lane = threadIdx.x & 31;
  int wave = blockIdx.x * 4 + wv;
  int b  = wave >> 7;
  int n0 = (wave & 127) << 4;
  int hl  = lane >> 4;
  int l16 = lane & 15;

  const float* Vb = V + (size_t)b * NN * EE;

  v2f a[4];
  const float* arow = Vb + (size_t)(n0 + l16) * EE;
#pragma unroll
  for (int kk = 0; kk < 4; ++kk) {
    int k0 = kk * 4 + hl * 2;
    a[kk].x = arow[k0]; a[kk].y = arow[k0 + 1];
  }

  v8f acc[4];
#pragma unroll
  for (int ct = 0; ct < 4; ++ct) acc[ct] = (v8f){};

  for (int m0 = 0; m0 < NN; m0 += 16) {
    // --- S tile = V_rows @ V_m^T ---
    const float* brow = Vb + (size_t)(m0 + l16) * EE;
    v8f c = {};
#pragma unroll
    for (int kk = 0; kk < 4; ++kk) {
      int k0 = kk * 4 + hl * 2;
      v2f bb; bb.x = brow[k0]; bb.y = brow[k0 + 1];
      c = wmma4(a[kk], bb, c);
    }
    float dm = dv[b * NN + m0 + l16];

    __syncthreads();                  // prior iteration's LDS reads complete
#pragma unroll
    for (int i = 0; i < 8; ++i)
      sbuf[wv][i + hl * 8][l16] = fmaxf(c[i], 0.f) * dm;   // C-layout store
    __syncthreads();

    v2f a2[4];                        // reload as A operand (rows n, K = m)
#pragma unroll
    for (int kk = 0; kk < 4; ++kk) {
      int k0 = kk * 4 + hl * 2;
      a2[kk].x = sbuf[wv][l16][k0];
      a2[kk].y = sbuf[wv][l16][k0 + 1];
    }

    // --- acc += S_scaled @ x_tile (16 x 64) ---
    const float* xb = x + ((size_t)b * NN + m0) * CC;
#pragma unroll
    for (int ct = 0; ct < 4; ++ct) {
      int o = ct * 16 + l16;
#pragma unroll
      for (int kk = 0; kk < 4; ++kk) {
        int k0 = kk * 4 + hl * 2;
        v2f bb;
        bb.x = xb[(size_t)k0 * CC + o];
        bb.y = xb[(size_t)(k0 + 1) * CC + o];
        acc[ct] = wmma4(a2[kk], bb, acc[ct]);
      }
    }
  }

  // scale rows by d[n] and store
#pragma unroll
  for (int i = 0; i < 8; ++i) {
    int row = n0 + i + hl * 8;
    float dn = dv[b * NN + row];
#pragma unroll
    for (int ct = 0; ct < 4; ++ct)
      xg2[((size_t)b * NN + row) * CC + ct * 16 + l16] = acc[ct][i] * dn;
  }
}

// ---------------------------------------------------------------------------
// Phase 4: per-node hyper-weights + final GEMM.
// Block = node n. A = xg[:, n, :, :] (16 batches x 128), B = W[n] (128 x 64)
// built chunk-wise in LDS from emb1[n] (uniform -> scalar loads).
// ---------------------------------------------------------------------------
__global__ void k_out(const float* __restrict__ x, const float* __restrict__ xg2,
                      const float* __restrict__ emb1, const float* __restrict__ wp,
                      const float* __restrict__ bp, float* __restrict__ out) {
  __shared__ float Ab[16][128];       // [batch][k*64+i]
  __shared__ float Wc[16][64];        // current 16-row chunk of W[n]

  int n   = blockIdx.x;
  int tid = threadIdx.x;              // 128 threads = 4 waves
  int lane = tid & 31, wv = tid >> 5, hl = lane >> 4, l16 = lane & 15;

  float emb[16];
#pragma unroll
  for (int d = 0; d < 16; ++d) emb[d] = emb1[n * 16 + d];   // uniform

  // stage A: cols 0..63 = x (k=0), cols 64..127 = xg2 (k=1)
#pragma unroll
  for (int j = 0; j < 16; ++j) {
    int idx = tid + j * 128;
    int bb  = idx >> 7;
    int col = idx & 127;
    float v = (col < 64)
                  ? x[((size_t)bb * NN + n) * CC + col]
                  : xg2[((size_t)bb * NN + n) * CC + (col - 64)];
    Ab[bb][col] = v;
  }
  __syncthreads();

  v8f acc = {};                       // each wave owns one 16-col output tile
  for (int kc = 0; kc < 8; ++kc) {
    // build W chunk: Wc[r][o] = sum_d emb[d] * wp[d, kc*16+r, o]
#pragma unroll
    for (int j = 0; j < 8; ++j) {
      int idx = tid + j * 128;
      int r = idx >> 6;
      int o = idx & 63;
      int ki = kc * 16 + r;
      float s = 0.f;
#pragma unroll
      for (int d = 0; d < 16; ++d)
        s = fmaf(emb[d], wp[d * 8192 + ki * 64 + o], s);
      Wc[r][o] = s;
    }
    __syncthreads();

#pragma unroll
    for (int kk = 0; kk < 4; ++kk) {
      int k0 = kk * 4 + hl * 2;
      v2f a; a.x = Ab[l16][kc * 16 + k0]; a.y = Ab[l16][kc * 16 + k0 + 1];
      int o = wv * 16 + l16;
      v2f b; b.x = Wc[k0][o]; b.y = Wc[k0 + 1][o];
      acc = wmma4(a, b, acc);
    }
    __syncthreads();
  }

  // bias[n,o] = emb1[n] @ bias_pool, then store out[b,n,o]
  int o = wv * 16 + l16;
  float bias = 0.f;
#pragma unroll
  for (int d = 0; d < 16; ++d) bias = fmaf(emb[d], bp[d * 64 + o], bias);

#pragma unroll
  for (int i = 0; i < 8; ++i) {
    int bb = i + hl * 8;
    out[((size_t)bb * NN + n) * OO + o] = acc[i] + bias;
  }
}

// ---------------------------------------------------------------------------
extern "C" void kernel_launch(void* const* d_in, const int* in_sizes, int n_in,
                              void* d_out, int out_size, void* d_ws, size_t ws_size,
                              hipStream_t stream) {
  const float* x    = (const float*)d_in[0];
  const float* emb0 = (const float*)d_in[1];
  const float* emb1 = (const float*)d_in[2];
  const float* w1   = (const float*)d_in[3];
  const float* b1   = (const float*)d_in[4];
  const float* w2   = (const float*)d_in[5];
  const float* b2   = (const float*)d_in[6];
  const float* w3   = (const float*)d_in[7];
  const float* b3   = (const float*)d_in[8];
  const float* wp   = (const float*)d_in[9];
  const float* bp   = (const float*)d_in[10];
  float* out = (float*)d_out;

  float* ws  = (float*)d_ws;
  float* V   = ws;                                  // B*N*E   = 524288 f
  float* dv  = V + (size_t)BATCH * NN * EE;         // B*N     = 32768 f
  float* xg2 = dv + (size_t)BATCH * NN;             // B*N*C   = 2097152 f

  k_hyper<<<(BATCH * NN) / 256, 256, 0, stream>>>(x, emb0, w1, b1, w2, b2, w3, b3, V);
  k_rowsum<<<(BATCH * NN / 16) / 4, 128, 0, stream>>>(V, dv);
  k_lx<<<(BATCH * NN / 16) / 4, 128, 0, stream>>>(V, dv, x, xg2);
  k_out<<<NN, 128, 0, stream>>>(x, xg2, emb1, wp, bp, out);
}